// LabelSmoothing_ocr_28166395527860
// MI455X (gfx1250) — compile-verified
//
#include <hip/hip_runtime.h>
#include <cstdint>

// ---------------- problem constants ----------------
#define BATCH 16
#define HH    1024
#define WW    1024
#define HW    (HH * WW)
#define TW    64          // tile width  (x)
#define TH    32          // tile height (y)
#define PADR  5           // (11-1)/2
#define NPIX  16777216.0  // BATCH*HH*WW

typedef float v2f __attribute__((ext_vector_type(2)));
typedef float v8f __attribute__((ext_vector_type(8)));

typedef __attribute__((address_space(1))) char* as1_cp;
typedef __attribute__((address_space(3))) char* as3_cp;

#if defined(__has_builtin)
#  if __has_builtin(__builtin_amdgcn_wmma_f32_16x16x4_f32)
#    define HAVE_WMMA44 1
#  endif
#  if __has_builtin(__builtin_amdgcn_global_load_async_to_lds_b8)
#    define HAVE_ASYNC_B8 1
#  endif
#endif

// ws layout: 5 doubles: [0]=sum_edge [1]=base0 [2]=lin0 [3]=base1 [4]=lin1
__global__ void zero_acc_kernel(double* acc) {
    if (threadIdx.x < 8) acc[threadIdx.x] = 0.0;
}

__global__ __launch_bounds__(256) void fused_loss_kernel(
    const float* __restrict__ score0,
    const float* __restrict__ score1,
    const int*   __restrict__ target,
    double*      __restrict__ acc)
{
    __shared__ unsigned char tloc[TH + 10][80];   // target tile + halo (74 used, padded)
    __shared__ float hsum[TH + 10][TW];           // horizontal 11-sums
    __shared__ float box[TH][TW];                 // 11x11 box sums (WMMA output)
    __shared__ float wred[8][5];                  // per-wave partials

    const int tid = threadIdx.x;
    const int b   = blockIdx.z;
    const int gx0 = blockIdx.x * TW;
    const int gy0 = blockIdx.y * TH;
    const int* tgt = target + (size_t)b * HW;

    // ---------------- Phase A: load target halo (74x42) into LDS ----------------
#if defined(HAVE_ASYNC_B8)
    // pre-zero (covers zero-padding for image borders + masked lanes)
    for (int idx = tid; idx < (TH + 10) * 80; idx += 256)
        ((unsigned char*)tloc)[idx] = 0;
    __syncthreads();
    for (int idx = tid; idx < (TH + 10) * (TW + 10); idx += 256) {
        const int r = idx / (TW + 10), c = idx % (TW + 10);
        const int y = gy0 - PADR + r,  x = gx0 - PADR + c;
        if (y >= 0 && y < HH && x >= 0 && x < WW) {
            // target values are 0/1; low byte of the little-endian int IS the value
            const void* gp = (const void*)(tgt + (size_t)y * WW + x);
            const unsigned int lp = (unsigned int)(uintptr_t)&tloc[r][c];
            __builtin_amdgcn_global_load_async_to_lds_b8(
                (as1_cp)(uintptr_t)gp,
                (as3_cp)lp,
                0, 0);
        }
    }
    asm volatile("s_wait_asynccnt 0" ::: "memory");
    __syncthreads();
#else
    for (int idx = tid; idx < (TH + 10) * (TW + 10); idx += 256) {
        const int r = idx / (TW + 10), c = idx % (TW + 10);
        const int y = gy0 - PADR + r,  x = gx0 - PADR + c;
        unsigned char v = 0;
        if (y >= 0 && y < HH && x >= 0 && x < WW)
            v = (unsigned char)(tgt[(size_t)y * WW + x] != 0);
        tloc[r][c] = v;
    }
    __syncthreads();
#endif

    // ---------------- Phase B: horizontal 11-sums ----------------
    for (int idx = tid; idx < (TH + 10) * TW; idx += 256) {
        const int r = idx / TW, x = idx % TW;
        int s = 0;
#pragma unroll
        for (int d = 0; d < 11; ++d) s += tloc[r][x + d];
        hsum[r][x] = (float)s;
    }
    __syncthreads();

    // ---------------- Phase C: vertical 11-sum as WMMA matmul ----------------
    // Box(16x16) = Lband(16x26) * Hsum(26x16); K-sliced into 7 x (16x16x4) f32 WMMAs.
    // All operands are small integers -> exact in f32.
#if defined(HAVE_WMMA44)
    {
        const int w    = tid >> 5;          // wave id 0..7 -> one 16x16 output tile
        const int lane = tid & 31;
        const int half = lane >> 4;         // 0 / 1
        const int n    = lane & 15;         // column within tile (B/D N index)
        const int M    = lane & 15;         // row of A held by this lane
        const int y0   = (w >> 2) * 16;     // 0 or 16
        const int x0   = (w & 3) * 16;      // 0,16,32,48
        v8f d = {};
#pragma unroll
        for (int s = 0; s < 7; ++s) {
            v2f a, bb;
#pragma unroll
            for (int e = 0; e < 2; ++e) {
                const int jj = 4 * s + 2 * half + e;          // global K index 0..27
                a[e]  = (jj >= M && jj <= M + 10) ? 1.0f : 0.0f;
                bb[e] = (jj < 26) ? hsum[y0 + jj][x0 + n] : 0.0f;
            }
            d = __builtin_amdgcn_wmma_f32_16x16x4_f32(
                    false, a, false, bb, (short)0, d, false, false);
        }
#pragma unroll
        for (int g = 0; g < 8; ++g)
            box[y0 + g + 8 * half][x0 + n] = d[g];
    }
    __syncthreads();
#endif

    // ---------------- Phase D: fused elementwise loss (memory-bound core) ----------------
    float aE = 0.f, aB0 = 0.f, aL0 = 0.f, aB1 = 0.f, aL1 = 0.f;
    const int tx = (tid & 15) * 4;   // float4 column group
    const int ty = tid >> 4;         // 0..15
#pragma unroll
    for (int rr = 0; rr < 2; ++rr) {
        const int ly = ty + rr * 16;
        const int gy = gy0 + ly;
        const size_t p0 = ((size_t)(b * 2 + 0) * HH + gy) * WW + gx0 + tx;
        const size_t p1 = p0 + (size_t)HW;   // channel 1
        const float4 s0a = *(const float4*)(score0 + p0);
        const float4 s0b = *(const float4*)(score0 + p1);
        const float4 s1a = *(const float4*)(score1 + p0);
        const float4 s1b = *(const float4*)(score1 + p1);
#pragma unroll
        for (int j = 0; j < 4; ++j) {
            const int lx = tx + j;
            const int tc = (int)tloc[ly + PADR][lx + PADR];
#if defined(HAVE_WMMA44)
            const float S = box[ly][lx];
#else
            float S = 0.f;
#pragma unroll
            for (int d = 0; d < 11; ++d) S += hsum[ly + d][lx];
#endif
            // edge = (121*tc - S) != 0
            const float e = (S == (tc ? 121.0f : 0.0f)) ? 0.0f : 1.0f;
            aE += e;
            {
                const float c0 = ((const float*)&s0a)[j];
                const float c1 = ((const float*)&s0b)[j];
                const float m   = fmaxf(c0, c1);
                const float lse = m + log1pf(__expf(-fabsf(c0 - c1)));
                const float xt = tc ? c1 : c0, xo = tc ? c0 : c1;
                aB0 += lse - xt;
                aL0 += e * (0.5f * (xt + lse) - xo);
            }
            {
                const float c0 = ((const float*)&s1a)[j];
                const float c1 = ((const float*)&s1b)[j];
                const float m   = fmaxf(c0, c1);
                const float lse = m + log1pf(__expf(-fabsf(c0 - c1)));
                const float xt = tc ? c1 : c0, xo = tc ? c0 : c1;
                aB1 += lse - xt;
                aL1 += e * (0.5f * (xt + lse) - xo);
            }
        }
    }

    // ---------------- reduction: wave32 shuffle -> LDS -> f64 atomics ----------------
    float v[5] = {aE, aB0, aL0, aB1, aL1};
#pragma unroll
    for (int off = 16; off > 0; off >>= 1) {
#pragma unroll
        for (int k = 0; k < 5; ++k) v[k] += __shfl_xor(v[k], off, 32);
    }
    if ((tid & 31) == 0) {
#pragma unroll
        for (int k = 0; k < 5; ++k) wred[tid >> 5][k] = v[k];
    }
    __syncthreads();
    if (tid < 5) {
        float s = 0.f;
#pragma unroll
        for (int w = 0; w < 8; ++w) s += wred[w][tid];
        atomicAdd(&acc[tid], (double)s);
    }
}

__global__ void finalize_kernel(const double* __restrict__ acc, float* __restrict__ out) {
    double a = acc[0] / NPIX;
    if (a > 0.2) a = 0.2;
    const double loss =
        (1.0 * (acc[1] + a * acc[2]) + 0.5 * (acc[3] + a * acc[4])) / NPIX;
    out[0] = (float)loss;
}

extern "C" void kernel_launch(void* const* d_in, const int* in_sizes, int n_in,
                              void* d_out, int out_size, void* d_ws, size_t ws_size,
                              hipStream_t stream) {
    const float* score0 = (const float*)d_in[0];
    const float* score1 = (const float*)d_in[1];
    const int*   target = (const int*)d_in[2];
    double* acc = (double*)d_ws;           // 40 bytes used
    (void)in_sizes; (void)n_in; (void)out_size; (void)ws_size;

    zero_acc_kernel<<<dim3(1), dim3(32), 0, stream>>>(acc);
    dim3 grid(WW / TW, HH / TH, BATCH);    // 16 x 32 x 16
    fused_loss_kernel<<<grid, dim3(256), 0, stream>>>(score0, score1, target, acc);
    finalize_kernel<<<dim3(1), dim3(1), 0, stream>>>(acc, (float*)d_out);
}